// GatConv_35192962024014
// MI455X (gfx1250) — compile-verified
//
#include <hip/hip_runtime.h>

#define NVAL 300
#define EVAL 100
#define NTILES 19   // ceil(300/16) i/j tiles
#define ETILES 7    // ceil(100/16) output col tiles
#define KSTEPS1 25  // E/4
#define KSTEPS2 75  // N/4  (300 divisible by 4)
#define NEG_INF -9.0e15f
#define SLOPE 0.2f
#define HROWS 304   // N padded to 16

typedef __attribute__((ext_vector_type(2))) float v2f;
typedef __attribute__((ext_vector_type(8))) float v8f;

#define LDS_H_FLOATS (HROWS * EVAL)   // 30400
#define LDS_ALPHA_FLOATS (16 * NVAL)  // 4800
#define LDS_A_FLOATS (4 * EVAL)       // 400

__global__ __launch_bounds__(256) void gat_wmma_kernel(
    const float* __restrict__ h,    // [B,300,100] f32
    const int* __restrict__ adj,    // [B,300,300] int32 (harness int convention)
    const float* __restrict__ Ac,   // [4,100] f32
    float* __restrict__ out)        // [B,300,100] f32
{
    __shared__ float sh[LDS_H_FLOATS + LDS_ALPHA_FLOATS + LDS_A_FLOATS];
    float* h_lds = sh;
    float* alpha_lds = sh + LDS_H_FLOATS;
    float* a_lds = alpha_lds + LDS_ALPHA_FLOATS;

    const int b    = blockIdx.y;
    const int i0   = blockIdx.x * 16;
    const int tid  = threadIdx.x;
    const int wave = tid >> 5;
    const int lane = tid & 31;
    const int half = lane >> 4;   // K half-split per ISA A/B layouts
    const int lpos = lane & 15;   // M (A) or N (B) position

    // ---- Stage h[b] (30000 f32, contiguous) into LDS as float4 ----
    {
        const float4* hsrc = (const float4*)(h + (size_t)b * (NVAL * EVAL));
        float4* hdst = (float4*)h_lds;
        for (int idx = tid; idx < (NVAL * EVAL) / 4; idx += 256)
            hdst[idx] = hsrc[idx];
        float4 z{0.f, 0.f, 0.f, 0.f};
        for (int idx = tid; idx < ((HROWS - NVAL) * EVAL) / 4; idx += 256)
            hdst[(NVAL * EVAL) / 4 + idx] = z;  // zero rows 300..303
        for (int idx = tid; idx < LDS_A_FLOATS / 4; idx += 256)
            ((float4*)a_lds)[idx] = ((const float4*)Ac)[idx];
    }
    __syncthreads();

    // ---- Pass 1: masked logits.  e_k[i,j] = sum_d (h[i,d]*A[k,d]) * h[j,d]
    // A-frag 16x4: lane m=lpos, K = v + 2*half. B-frag 4x16: lane n=lpos, same K split.
    {
        const float* arow = h_lds + (i0 + lpos) * EVAL;
        for (int jt = wave; jt < NTILES; jt += 8) {
            const int j0 = jt * 16;
            const float* brow = h_lds + (j0 + lpos) * EVAL;
            v8f D0 = {}, D1 = {}, D2 = {}, D3 = {};
#pragma unroll 5
            for (int s = 0; s < KSTEPS1; ++s) {
                const int d = 4 * s + 2 * half;
                v2f ha  = *(const v2f*)(arow + d);
                v2f hb  = *(const v2f*)(brow + d);
                v2f c0  = *(const v2f*)(a_lds + 0 * EVAL + d);
                v2f c1  = *(const v2f*)(a_lds + 1 * EVAL + d);
                v2f c2  = *(const v2f*)(a_lds + 2 * EVAL + d);
                v2f c3  = *(const v2f*)(a_lds + 3 * EVAL + d);
                v2f f0 = ha * c0, f1 = ha * c1, f2 = ha * c2, f3 = ha * c3;
                D0 = __builtin_amdgcn_wmma_f32_16x16x4_f32(false, f0, false, hb, (short)0, D0, false, false);
                D1 = __builtin_amdgcn_wmma_f32_16x16x4_f32(false, f1, false, hb, (short)0, D1, false, false);
                D2 = __builtin_amdgcn_wmma_f32_16x16x4_f32(false, f2, false, hb, (short)0, D2, false, false);
                D3 = __builtin_amdgcn_wmma_f32_16x16x4_f32(false, f3, false, hb, (short)0, D3, false, false);
            }
            // lrelu + adj select + store masked logits (C/D: vgpr r -> M = r + 8*half, N = lpos)
            const int gj = j0 + lpos;
            if (gj < NVAL) {
#pragma unroll
                for (int r = 0; r < 8; ++r) {
                    const int il = r + 8 * half;
                    const int gi = i0 + il;
                    int a = 0;
                    if (gi < NVAL)
                        a = adj[((size_t)b * NVAL + gi) * NVAL + gj];
                    float e0 = D0[r]; e0 = e0 >= 0.f ? e0 : SLOPE * e0;
                    float e1 = D1[r]; e1 = e1 >= 0.f ? e1 : SLOPE * e1;
                    float e2 = D2[r]; e2 = e2 >= 0.f ? e2 : SLOPE * e2;
                    float e3 = D3[r]; e3 = e3 >= 0.f ? e3 : SLOPE * e3;
                    float v = NEG_INF;
                    if (a == 1) v = e0;
                    else if (a == 2) v = e1;
                    else if (a == 3) v = e2;
                    else if (a == 4) v = e3;
                    alpha_lds[il * NVAL + gj] = v;
                }
            }
        }
    }
    __syncthreads();

    // ---- Pass 2: row softmax over j (16 rows, 2 per wave), wave32 shuffles ----
    for (int row = wave * 2; row < wave * 2 + 2; ++row) {
        float* rp = alpha_lds + row * NVAL;
        float m = -INFINITY;
        for (int j = lane; j < NVAL; j += 32) m = fmaxf(m, rp[j]);
#pragma unroll
        for (int off = 16; off; off >>= 1) m = fmaxf(m, __shfl_xor(m, off));
        float s = 0.f;
        for (int j = lane; j < NVAL; j += 32) {
            float ex = __expf(rp[j] - m);
            rp[j] = ex;
            s += ex;
        }
#pragma unroll
        for (int off = 16; off; off >>= 1) s += __shfl_xor(s, off);
        const float inv = 1.0f / s;
        for (int j = lane; j < NVAL; j += 32) rp[j] *= inv;
    }
    __syncthreads();

    // ---- Pass 3: out[16,100] = alpha[16,300] @ h[300,100] via f32 WMMA ----
    for (int et = wave; et < ETILES; et += 8) {
        const int e0 = et * 16;
        const float* arow = alpha_lds + lpos * NVAL;
        v8f D = {};
#pragma unroll 5
        for (int s = 0; s < KSTEPS2; ++s) {
            const int d = 4 * s + 2 * half;     // K base for this lane half
            v2f af = *(const v2f*)(arow + d);
            v2f bf = { h_lds[(d + 0) * EVAL + e0 + lpos],
                       h_lds[(d + 1) * EVAL + e0 + lpos] };
            D = __builtin_amdgcn_wmma_f32_16x16x4_f32(false, af, false, bf, (short)0, D, false, false);
        }
        const int ge = e0 + lpos;
        if (ge < EVAL) {
#pragma unroll
            for (int r = 0; r < 8; ++r) {
                const int gi = i0 + r + 8 * half;
                if (gi < NVAL)
                    out[((size_t)b * NVAL + gi) * EVAL + ge] = D[r];
            }
        }
    }
}

extern "C" void kernel_launch(void* const* d_in, const int* in_sizes, int n_in,
                              void* d_out, int out_size, void* d_ws, size_t ws_size,
                              hipStream_t stream) {
    const float* h  = (const float*)d_in[0];
    const int* adj  = (const int*)d_in[1];
    const float* Ac = (const float*)d_in[2];
    float* out = (float*)d_out;
    const int B = in_sizes[0] / (NVAL * EVAL);  // 256
    dim3 grid(NTILES, B);
    gat_wmma_kernel<<<grid, 256, 0, stream>>>(h, adj, Ac, out);
}